// RelativeEventEmbedding_5549097746626
// MI455X (gfx1250) — compile-verified
//
#include <hip/hip_runtime.h>
#include <math.h>

#define BB 8
#define SS 2048
#define FD 64
#define DM 512
#define NSC 5

typedef __attribute__((ext_vector_type(2))) float v2f;
typedef __attribute__((ext_vector_type(8))) float v8f;

// ---------------------------------------------------------------- WMMA helpers

__device__ __forceinline__ v8f wmma_f32(v2f a, v2f b, v8f c) {
  // V_WMMA_F32_16X16X4_F32 : D = A(16x4) * B(4x16) + C(16x16)
  return __builtin_amdgcn_wmma_f32_16x16x4_f32(false, a, false, b, (short)0, c,
                                               false, false);
}

// Universal fragment loader for both A (16xK rows) and B (KxN = rows of the
// transposed operand) per CDNA5 32-bit layouts:
//   lane l: i = l&15, h = l>>4 ; {elem(i, kb+2h), elem(i, kb+2h+1)}
__device__ __forceinline__ v2f ldfrag(const float* p, int ld, int row0, int kb,
                                      int lane) {
  int n = lane & 15, h = lane >> 4;
  const float* q = p + (size_t)(row0 + n) * ld + (kb + 2 * h);
  v2f r;
  r.x = q[0];
  r.y = q[1];
  return r;
}

// monotone float <-> uint keys for atomic min/max (handles negatives)
__device__ __forceinline__ unsigned fenc(float f) {
  unsigned u = __float_as_uint(f);
  return (u & 0x80000000u) ? ~u : (u | 0x80000000u);
}
__device__ __forceinline__ float fdec(unsigned k) {
  return __uint_as_float((k & 0x80000000u) ? (k & 0x7FFFFFFFu) : ~k);
}

// ---------------------------------------------------------------- K0: init

__global__ void k_init(double* dsum, unsigned* keys) {
  if (threadIdx.x == 0) {
    dsum[0] = 0.0;
    dsum[1] = 0.0;
    keys[0] = 0xFFFFFFFFu;  // min key
    keys[1] = 0x00000000u;  // max key
  }
}

// ------------------------------------------------- K1: q,k projection + norms

__global__ void k_rows(const float* __restrict__ x, const float* __restrict__ wq,
                       const float* __restrict__ bq, const float* __restrict__ wk,
                       const float* __restrict__ bk, float* __restrict__ qb,
                       float* __restrict__ kb, float* __restrict__ sq,
                       float* __restrict__ invn) {
  int wid = (blockIdx.x * blockDim.x + threadIdx.x) >> 5;
  int lane = threadIdx.x & 31;
  int b = wid / (SS / 16);
  int s0 = (wid % (SS / 16)) * 16;
  const float* xb = x + (size_t)b * SS * FD;
  int n = lane & 15, h = lane >> 4;

  v2f ax[16];
#pragma unroll
  for (int kk = 0; kk < 16; ++kk) ax[kk] = ldfrag(xb, FD, s0, 4 * kk, lane);

#pragma unroll
  for (int nb = 0; nb < 4; ++nb) {
    // prefetch weight fragments into distinct registers, then run the chains
    v2f wfq[16], wfk[16];
#pragma unroll
    for (int kk = 0; kk < 16; ++kk) {
      wfq[kk] = ldfrag(wq, FD, nb * 16, 4 * kk, lane);
      wfk[kk] = ldfrag(wk, FD, nb * 16, 4 * kk, lane);
    }
    v8f cq = {0, 0, 0, 0, 0, 0, 0, 0};
    v8f ck = {0, 0, 0, 0, 0, 0, 0, 0};
#pragma unroll
    for (int kk = 0; kk < 16; ++kk) {
      cq = wmma_f32(ax[kk], wfq[kk], cq);
      ck = wmma_f32(ax[kk], wfk[kk], ck);
    }
    float biq = bq[nb * 16 + n], bik = bk[nb * 16 + n];
#pragma unroll
    for (int v = 0; v < 8; ++v) {
      size_t r = (size_t)b * SS + s0 + v + 8 * h;
      qb[r * FD + nb * 16 + n] = cq[v] + biq;
      kb[r * FD + nb * 16 + n] = ck[v] + bik;
    }
  }
  // sum of squares + inverse norm per row (two half-waves split the 64 cols)
  const float* xr = xb + (size_t)(s0 + n) * FD + h * 32;
  float p = 0.f;
#pragma unroll
  for (int f = 0; f < 32; ++f) p += xr[f] * xr[f];
  p += __shfl_xor(p, 16, 32);
  if (h == 0) {
    sq[(size_t)b * SS + s0 + n] = p;
    invn[(size_t)b * SS + s0 + n] = 1.f / fmaxf(sqrtf(p), 1e-12f);
  }
}

// ----------------------------------------- K2: euclid sum / sumsq (WMMA Gram)

__global__ void k_stats(const float* __restrict__ x, const float* __restrict__ sq,
                        double* __restrict__ dsum) {
  int wid = (blockIdx.x * blockDim.x + threadIdx.x) >> 5;
  int lane = threadIdx.x & 31;
  int n = lane & 15, h = lane >> 4;
  int b = wid / (SS / 16);
  int s0 = (wid % (SS / 16)) * 16;
  const float* xb = x + (size_t)b * SS * FD;
  const float* sqb = sq + (size_t)b * SS;

  v2f ax[16];
#pragma unroll
  for (int kk = 0; kk < 16; ++kk) ax[kk] = ldfrag(xb, FD, s0, 4 * kk, lane);
  float sqs[8];
#pragma unroll
  for (int v = 0; v < 8; ++v) sqs[v] = sqb[s0 + v + 8 * h];

  double lsum = 0.0, lsq2 = 0.0;
  for (int t0 = 0; t0 < SS; t0 += 16) {
    float sqt = sqb[t0 + n];
    v2f bf[16];
#pragma unroll
    for (int kk = 0; kk < 16; ++kk) bf[kk] = ldfrag(xb, FD, t0, 4 * kk, lane);
    v8f cg = {0, 0, 0, 0, 0, 0, 0, 0};
#pragma unroll
    for (int kk = 0; kk < 16; ++kk) cg = wmma_f32(ax[kk], bf[kk], cg);
    float ps = 0.f, pq = 0.f;
#pragma unroll
    for (int v = 0; v < 8; ++v) {
      float d2 = fmaxf(sqs[v] + sqt - 2.f * cg[v], 0.f);
      float eu = (d2 > 0.f) ? sqrtf(fmaxf(d2, 1e-12f)) : 0.f;
      ps += eu;
      pq += eu * eu;
    }
    lsum += (double)ps;
    lsq2 += (double)pq;
  }
#pragma unroll
  for (int msk = 1; msk < 32; msk <<= 1) {
    lsum += __shfl_xor(lsum, msk, 32);
    lsq2 += __shfl_xor(lsq2, msk, 32);
  }
  if (lane == 0) {
    atomicAdd(&dsum[0], lsum);
    atomicAdd(&dsum[1], lsq2);
  }
}

// ------------------------------------------------- K3: scalar constants pt. 1

__global__ void k_scal1(const double* __restrict__ dsum,
                        const float* __restrict__ alpha,
                        const float* __restrict__ sigw,
                        float* __restrict__ consts) {
  if (threadIdx.x == 0) {
    double N = (double)BB * (double)SS * (double)SS;
    double mu = dsum[0] / N;
    double var = (dsum[1] - N * mu * mu) / (N - 1.0);
    float stdv = (float)sqrt(fmax(var, 0.0));
    consts[0] = (float)mu;      // mean
    consts[1] = 2.f * stdv;     // clip threshold
    consts[4] = 1.f / (1.f + __expf(-alpha[0]));  // sigmoid(alpha)
    float swv = sigw[0];
    float sp = (swv > 20.f) ? swv : log1pf(__expf(swv));
    float sig = sp + 0.001f;
    consts[5] = 1.f / (2.f * sig * sig);          // 1/(2 sigma^2)
  }
}

// ----------------------------------- K4: clipped euclid min/max (WMMA Gram)

__global__ void k_minmax(const float* __restrict__ x,
                         const float* __restrict__ sq,
                         const float* __restrict__ consts,
                         unsigned* __restrict__ keys) {
  int wid = (blockIdx.x * blockDim.x + threadIdx.x) >> 5;
  int lane = threadIdx.x & 31;
  int n = lane & 15, h = lane >> 4;
  int b = wid / (SS / 16);
  int s0 = (wid % (SS / 16)) * 16;
  const float* xb = x + (size_t)b * SS * FD;
  const float* sqb = sq + (size_t)b * SS;
  float mean = consts[0], thr = consts[1];

  v2f ax[16];
#pragma unroll
  for (int kk = 0; kk < 16; ++kk) ax[kk] = ldfrag(xb, FD, s0, 4 * kk, lane);
  float sqs[8];
#pragma unroll
  for (int v = 0; v < 8; ++v) sqs[v] = sqb[s0 + v + 8 * h];

  float lmin = 1e30f, lmax = -1e30f;
  for (int t0 = 0; t0 < SS; t0 += 16) {
    float sqt = sqb[t0 + n];
    v2f bf[16];
#pragma unroll
    for (int kk = 0; kk < 16; ++kk) bf[kk] = ldfrag(xb, FD, t0, 4 * kk, lane);
    v8f cg = {0, 0, 0, 0, 0, 0, 0, 0};
#pragma unroll
    for (int kk = 0; kk < 16; ++kk) cg = wmma_f32(ax[kk], bf[kk], cg);
#pragma unroll
    for (int v = 0; v < 8; ++v) {
      float d2 = fmaxf(sqs[v] + sqt - 2.f * cg[v], 0.f);
      float eu = (d2 > 0.f) ? sqrtf(fmaxf(d2, 1e-12f)) : 0.f;
      float dev = eu - mean;
      eu = (fabsf(dev) > thr) ? (copysignf(thr, dev) + mean) : eu;
      lmin = fminf(lmin, eu);
      lmax = fmaxf(lmax, eu);
    }
  }
#pragma unroll
  for (int msk = 1; msk < 32; msk <<= 1) {
    lmin = fminf(lmin, __shfl_xor(lmin, msk, 32));
    lmax = fmaxf(lmax, __shfl_xor(lmax, msk, 32));
  }
  if (lane == 0) {
    atomicMin(&keys[0], fenc(lmin));
    atomicMax(&keys[1], fenc(lmax));
  }
}

// ------------------------------------------------- K5: scalar constants pt. 2

__global__ void k_scal2(const unsigned* __restrict__ keys,
                        float* __restrict__ consts) {
  if (threadIdx.x == 0) {
    float mn = fdec(keys[0]);
    float mx = fdec(keys[1]);
    consts[2] = mn;
    consts[3] = 1.f / (mx - mn + 1e-6f);
  }
}

// ---------- K6: fused Gram + scores + online softmax + gated E*x (flash pass)

__global__ void k_flash(const float* __restrict__ x, const float* __restrict__ qb,
                        const float* __restrict__ kb, const float* __restrict__ sq,
                        const float* __restrict__ invn,
                        const float* __restrict__ consts,
                        float* __restrict__ pe) {
  __shared__ float elds[4][16 * 17];
  int wid = (blockIdx.x * blockDim.x + threadIdx.x) >> 5;
  int wib = threadIdx.x >> 5;
  int lane = threadIdx.x & 31;
  int n = lane & 15, h = lane >> 4;
  int b = wid / (SS / 16);
  int s0 = (wid % (SS / 16)) * 16;
  const float* xb = x + (size_t)b * SS * FD;
  const float* qrow = qb + (size_t)b * SS * FD;
  const float* krow = kb + (size_t)b * SS * FD;
  const float* sqb = sq + (size_t)b * SS;
  const float* inb = invn + (size_t)b * SS;
  float mean = consts[0], thr = consts[1], minv = consts[2], rscale = consts[3];
  float asig = consts[4], i2s2 = consts[5];

  v2f ax[16], aq[16];
#pragma unroll
  for (int kk = 0; kk < 16; ++kk) {
    ax[kk] = ldfrag(xb, FD, s0, 4 * kk, lane);
    aq[kk] = ldfrag(qrow, FD, s0, 4 * kk, lane);
  }
  float sqs[8], ins[8], mrow[8], zrow[8];
#pragma unroll
  for (int v = 0; v < 8; ++v) {
    sqs[v] = sqb[s0 + v + 8 * h];
    ins[v] = inb[s0 + v + 8 * h];
    mrow[v] = -1e30f;
    zrow[v] = 0.f;
  }
  v8f acc[4];
#pragma unroll
  for (int nb = 0; nb < 4; ++nb) acc[nb] = (v8f){0, 0, 0, 0, 0, 0, 0, 0};
  float* my = &elds[wib][0];

  for (int t0 = 0; t0 < SS; t0 += 16) {
    float sqt = sqb[t0 + n];
    float ivt = inb[t0 + n];
    // ---- prefetch all B fragments for the two score chains (distinct regs)
    v2f bf[16], kf[16];
#pragma unroll
    for (int kk = 0; kk < 16; ++kk) {
      bf[kk] = ldfrag(xb, FD, t0, 4 * kk, lane);
      kf[kk] = ldfrag(krow, FD, t0, 4 * kk, lane);
    }
    v8f cg = {0, 0, 0, 0, 0, 0, 0, 0};
    v8f cs = {0, 0, 0, 0, 0, 0, 0, 0};
#pragma unroll
    for (int kk = 0; kk < 16; ++kk) {
      cg = wmma_f32(ax[kk], bf[kk], cg);
      cs = wmma_f32(aq[kk], kf[kk], cs);
    }
    // ---- prefetch PV B-operand fragments early (hidden under softmax VALU)
    v2f pv[16];
#pragma unroll
    for (int nb = 0; nb < 4; ++nb) {
#pragma unroll
      for (int kk = 0; kk < 4; ++kk) {
        int r = t0 + 4 * kk + 2 * h;
        v2f bx;
        bx.x = xb[(size_t)r * FD + nb * 16 + n];
        bx.y = xb[(size_t)(r + 1) * FD + nb * 16 + n];
        pv[nb * 4 + kk] = bx;
      }
    }
    float cf[8];
#pragma unroll
    for (int v = 0; v < 8; ++v) {
      float g = cg[v];
      float d2 = fmaxf(sqs[v] + sqt - 2.f * g, 0.f);
      float eu = (d2 > 0.f) ? sqrtf(fmaxf(d2, 1e-12f)) : 0.f;
      float dev = eu - mean;
      eu = (fabsf(dev) > thr) ? (copysignf(thr, dev) + mean) : eu;
      float en = (eu - minv) * rscale;
      float cosv = (g * ins[v] * ivt + 1.f) * 0.5f;
      float hyb = asig * cosv + (1.f - asig) * (1.f - en);
      float gate = __expf(-hyb * i2s2);
      float s = cs[v] * 0.125f;  // 1/sqrt(64)
      float rm = s;
#pragma unroll
      for (int msk = 1; msk < 16; msk <<= 1)
        rm = fmaxf(rm, __shfl_xor(rm, msk, 32));
      float mn = fmaxf(mrow[v], rm);
      float c = __expf(mrow[v] - mn);
      float e = __expf(s - mn) * gate;
      float rs = e;
#pragma unroll
      for (int msk = 1; msk < 16; msk <<= 1) rs += __shfl_xor(rs, msk, 32);
      zrow[v] = zrow[v] * c + rs;
      mrow[v] = mn;
      cf[v] = c;
      my[(v + 8 * h) * 17 + n] = e;  // E tile -> LDS in (row, col) form
    }
    // rescale the P accumulators by per-row correction factors
#pragma unroll
    for (int nb = 0; nb < 4; ++nb) {
#pragma unroll
      for (int v = 0; v < 8; ++v) acc[nb][v] *= cf[v];
    }
    asm volatile("s_wait_dscnt 0" ::: "memory");  // LDS stores visible (same wave)
    // reload E as A-matrix fragments (K=16 -> 4 k-steps)
    v2f ea[4];
#pragma unroll
    for (int kk = 0; kk < 4; ++kk) {
      int col = 4 * kk + 2 * h;
      v2f t;
      t.x = my[n * 17 + col];
      t.y = my[n * 17 + col + 1];
      ea[kk] = t;
    }
    // P(16x64) += E(16x16) * X[t0..t0+15](16x64)
#pragma unroll
    for (int nb = 0; nb < 4; ++nb) {
#pragma unroll
      for (int kk = 0; kk < 4; ++kk)
        acc[nb] = wmma_f32(ea[kk], pv[nb * 4 + kk], acc[nb]);
    }
  }
#pragma unroll
  for (int nb = 0; nb < 4; ++nb) {
#pragma unroll
    for (int v = 0; v < 8; ++v) {
      pe[((size_t)b * SS + s0 + v + 8 * h) * FD + nb * 16 + n] =
          acc[nb][v] / zrow[v];
    }
  }
}

// --------------------------------------- K7: pe_event * w_proj^T + b_proj

__global__ void k_proj(const float* __restrict__ pe,
                       const float* __restrict__ w_proj,
                       const float* __restrict__ b_proj,
                       float* __restrict__ out) {
  int wid = (blockIdx.x * blockDim.x + threadIdx.x) >> 5;
  int lane = threadIdx.x & 31;
  int n = lane & 15, h = lane >> 4;
  int b = wid / (SS / 16);
  int s0 = (wid % (SS / 16)) * 16;
  const float* pb = pe + (size_t)b * SS * FD;

  v2f ap[16];
#pragma unroll
  for (int kk = 0; kk < 16; ++kk) ap[kk] = ldfrag(pb, FD, s0, 4 * kk, lane);

  for (int nb = 0; nb < DM / 16; ++nb) {
    v2f wf[16];
#pragma unroll
    for (int kk = 0; kk < 16; ++kk)
      wf[kk] = ldfrag(w_proj, FD, nb * 16, 4 * kk, lane);
    v8f c = {0, 0, 0, 0, 0, 0, 0, 0};
#pragma unroll
    for (int kk = 0; kk < 16; ++kk) c = wmma_f32(ap[kk], wf[kk], c);
    float bias = b_proj[nb * 16 + n];
#pragma unroll
    for (int v = 0; v < 8; ++v)
      out[((size_t)b * SS + s0 + v + 8 * h) * DM + nb * 16 + n] = c[v] + bias;
  }
}

// -------------------- K8: rel-encoding + two layernorms + residual combine

__device__ __forceinline__ float blk_sum(float v, float* red) {
  int t = threadIdx.x;
  red[t] = v;
  __syncthreads();
  for (int s = 128; s > 0; s >>= 1) {
    if (t < s) red[t] += red[t + s];
    __syncthreads();
  }
  float r = red[0];
  __syncthreads();
  return r;
}

__global__ void k_final(const float* __restrict__ x,
                        const float* __restrict__ w_rel,
                        const float* __restrict__ b_rel,
                        const float* __restrict__ g1, const float* __restrict__ bt1,
                        const float* __restrict__ g2, const float* __restrict__ bt2,
                        const float* __restrict__ scale_w,
                        float* __restrict__ out) {
  __shared__ float red[256];
  int row = blockIdx.x;
  int b = row / SS, s = row % SS;
  int t = threadIdx.x;
  const float* xb = x + (size_t)b * SS * FD;
  float d_s = xb[(size_t)s * FD + 0];
  float e_s = xb[(size_t)s * FD + 1];
  float r0 = 0.f, r1 = 0.f, r2 = 0.f;
#pragma unroll
  for (int sc = 1; sc <= NSC; ++sc) {
    if (s >= sc) {
      float dd = xb[(size_t)(s - sc) * FD + 0] - d_s;
      float de = xb[(size_t)(s - sc) * FD + 1] - e_s;
      r0 += dd;
      r1 += de;
      r2 += scale_w[sc - 1] * __expf(-fabsf(dd) / (float)(1 << sc));
    }
  }
  float relv[2], prv[2];
  float sr = 0.f, sr2 = 0.f, sp = 0.f, sp2 = 0.f;
#pragma unroll
  for (int i = 0; i < 2; ++i) {
    int d = t + 256 * i;
    float rv = r0 * w_rel[d * 3 + 0] + r1 * w_rel[d * 3 + 1] +
               r2 * w_rel[d * 3 + 2] + 5.f * b_rel[d];
    float pv = out[(size_t)row * DM + d];
    relv[i] = rv;
    prv[i] = pv;
    sr += rv;
    sr2 += rv * rv;
    sp += pv;
    sp2 += pv * pv;
  }
  sr = blk_sum(sr, red);
  sr2 = blk_sum(sr2, red);
  sp = blk_sum(sp, red);
  sp2 = blk_sum(sp2, red);
  float mu1 = sr / DM;
  float rs1 = rsqrtf(fmaxf(sr2 / DM - mu1 * mu1, 0.f) + 1e-5f);
  float mu2 = sp / DM;
  float rs2 = rsqrtf(fmaxf(sp2 / DM - mu2 * mu2, 0.f) + 1e-5f);
#pragma unroll
  for (int i = 0; i < 2; ++i) {
    int d = t + 256 * i;
    out[(size_t)row * DM + d] = ((relv[i] - mu1) * rs1 * g1[d] + bt1[d]) +
                                ((prv[i] - mu2) * rs2 * g2[d] + bt2[d]);
  }
}

// ---------------------------------------------------------------- launcher

extern "C" void kernel_launch(void* const* d_in, const int* in_sizes, int n_in,
                              void* d_out, int out_size, void* d_ws,
                              size_t ws_size, hipStream_t stream) {
  const float* x = (const float*)d_in[0];
  const float* w_rel = (const float*)d_in[1];
  const float* b_rel = (const float*)d_in[2];
  const float* w_proj = (const float*)d_in[3];
  const float* b_proj = (const float*)d_in[4];
  const float* g1 = (const float*)d_in[5];
  const float* bt1 = (const float*)d_in[6];
  const float* g2 = (const float*)d_in[7];
  const float* bt2 = (const float*)d_in[8];
  const float* alpha = (const float*)d_in[9];
  const float* sigma_w = (const float*)d_in[10];
  const float* wq = (const float*)d_in[11];
  const float* bq = (const float*)d_in[12];
  const float* wk = (const float*)d_in[13];
  const float* bk = (const float*)d_in[14];
  const float* scale_w = (const float*)d_in[15];
  float* out = (float*)d_out;

  char* ws = (char*)d_ws;
  double* dsum = (double*)ws;               // [0]=sum, [1]=sumsq
  unsigned* keys = (unsigned*)(ws + 16);    // [0]=min key, [1]=max key
  float* consts = (float*)(ws + 32);        // mean,thr,minv,rscale,a,1/(2s^2)
  float* sq = (float*)(ws + 256);
  float* invn = sq + (size_t)BB * SS;
  float* qb = invn + (size_t)BB * SS;
  float* kb = qb + (size_t)BB * SS * FD;
  float* pe = kb + (size_t)BB * SS * FD;

  dim3 blk(128);                       // 4 waves/block (wave32)
  dim3 grd((BB * SS / 16) / 4);        // 1024 row-tiles total

  k_init<<<dim3(1), dim3(32), 0, stream>>>(dsum, keys);
  k_rows<<<grd, blk, 0, stream>>>(x, wq, bq, wk, bk, qb, kb, sq, invn);
  k_stats<<<grd, blk, 0, stream>>>(x, sq, dsum);
  k_scal1<<<dim3(1), dim3(32), 0, stream>>>(dsum, alpha, sigma_w, consts);
  k_minmax<<<grd, blk, 0, stream>>>(x, sq, consts, keys);
  k_scal2<<<dim3(1), dim3(32), 0, stream>>>(keys, consts);
  k_flash<<<grd, blk, 0, stream>>>(x, qb, kb, sq, invn, consts, pe);
  k_proj<<<grd, blk, 0, stream>>>(pe, w_proj, b_proj, out);
  k_final<<<dim3(BB * SS), dim3(256), 0, stream>>>(x, w_rel, b_rel, g1, bt1,
                                                   g2, bt2, scale_w, out);
}